// LFA_72464688218277
// MI455X (gfx1250) — compile-verified
//
#include <hip/hip_runtime.h>
#include <math.h>

// ---- problem constants (B=4, N=20480, K=16) ----
#define BB 4
#define NN 20480
#define KK 16
#define NT (NN / 16) // 16-point tiles per batch = 1280

typedef __attribute__((ext_vector_type(2))) float v2f;
typedef __attribute__((ext_vector_type(8))) float v8f;

__device__ __forceinline__ v8f zero8() {
  v8f z = {0.f, 0.f, 0.f, 0.f, 0.f, 0.f, 0.f, 0.f};
  return z;
}

// D(16x16,f32) += A(16x4,f32) * B(4x16,f32)  -- CDNA5 V_WMMA_F32_16X16X4_F32
__device__ __forceinline__ v8f wmma4(v2f a, v2f b, v8f c) {
  return __builtin_amdgcn_wmma_f32_16x16x4_f32(false, a, false, b, (short)0, c,
                                               false, false);
}

// A-matrix 16x4 f32 layout: lane L holds A[M=L%16, K=c0+(L<16?0:2)] in .x and
// K=c0+(L<16?1:3) in .y. K columns beyond ncols padded with 0.
__device__ __forceinline__ v2f load_A(const float* __restrict__ W, int ncols,
                                      int row0, int c0, int lane) {
  int row = row0 + (lane & 15);
  int cb = c0 + ((lane >> 4) << 1);
  v2f a;
  a.x = (cb < ncols) ? W[row * ncols + cb] : 0.0f;
  a.y = (cb + 1 < ncols) ? W[row * ncols + cb + 1] : 0.0f;
  return a;
}

// B-matrix 4x16 f32 from an LDS [chan][16] tile (rows = channels, padded rows
// must be zeroed by the producer).
__device__ __forceinline__ v2f load_B_lds(const float* T, int c0, int lane) {
  int col = lane & 15;
  int rb = c0 + ((lane >> 4) << 1);
  v2f b;
  b.x = T[rb * 16 + col];
  b.y = T[(rb + 1) * 16 + col];
  return b;
}

// B-matrix 4x16 f32 straight from a channel-major [C][N] global array.
__device__ __forceinline__ v2f load_B_chan(const float* __restrict__ X, int c0,
                                           int n, int lane) {
  int rb = c0 + ((lane >> 4) << 1);
  v2f b;
  b.x = X[rb * NN + n];
  b.y = X[(rb + 1) * NN + n];
  return b;
}

// wave32: reductions over the 16 "neighbor" lanes of each half-wave
__device__ __forceinline__ float hsum16(float v) {
#pragma unroll
  for (int m = 1; m < 16; m <<= 1) v += __shfl_xor(v, m, 16);
  return v;
}
__device__ __forceinline__ float hmax16(float v) {
#pragma unroll
  for (int m = 1; m < 16; m <<= 1) v = fmaxf(v, __shfl_xor(v, m, 16));
  return v;
}

// ---------------- Kernel 1: f_pc = cbr(m1_W @ feature) ----------------
__global__ void k1_fpc(const float* __restrict__ feature,
                       const float* __restrict__ m1W,
                       const float* __restrict__ m1g,
                       const float* __restrict__ m1b,
                       float* __restrict__ fpc) {
  int lane = threadIdx.x & 31;
  int wave = threadIdx.x >> 5;
  int tile = blockIdx.x * 8 + wave;
  int b = tile / NT;
  int n0 = (tile % NT) * 16;
  int col = lane & 15;
  int hb = (lane >> 4) * 8;
  const float* fb = feature + b * 16 * NN;
  v8f acc = zero8();
#pragma unroll
  for (int c0 = 0; c0 < 16; c0 += 4)
    acc = wmma4(load_A(m1W, 16, 0, c0, lane),
                load_B_chan(fb, c0, n0 + col, lane), acc);
#pragma unroll
  for (int r = 0; r < 8; ++r) {
    int o = r + hb;
    fpc[(b * 16 + o) * NN + n0 + col] =
        fmaxf(m1g[o] * acc[r] + m1b[o], 0.0f);
  }
}

// Geometry + lrep = cbr(lm1 @ local_rep) for one point; wave-per-point,
// k = lane&15 (both halves duplicate the neighborhood so EXEC stays full).
__device__ __forceinline__ v8f geo_lrep(
    const float* __restrict__ xyz, const int* __restrict__ nidx,
    const float* __restrict__ lm1W, const float* __restrict__ lm1g,
    const float* __restrict__ lm1b, int b, int n, float* ldsT /*12x16*/,
    int lane, float& gdis, int& idx, float& lg) {
  int k = lane & 15;
  idx = nidx[(b * NN + n) * KK + k];
  float px = xyz[(b * NN + n) * 3 + 0];
  float py = xyz[(b * NN + n) * 3 + 1];
  float pz = xyz[(b * NN + n) * 3 + 2];
  float qx = xyz[(b * NN + idx) * 3 + 0];
  float qy = xyz[(b * NN + idx) * 3 + 1];
  float qz = xyz[(b * NN + idx) * 3 + 2];
  float rx = px - qx, ry = py - qy, rz = pz - qz;
  float r_alpha = atan2f(ry, rx);
  float r_beta = atan2f(rz, sqrtf(rx * rx + ry * ry));
  float r_dis = sqrtf(rx * rx + ry * ry + rz * rz);
  gdis = __expf(-r_dis);
  float mx = hsum16(qx) * 0.0625f;
  float my = hsum16(qy) * 0.0625f;
  float mz = hsum16(qz) * 0.0625f;
  float dx = px - mx, dy = py - my, dz = pz - mz;
  float da = atan2f(dy, dx);
  float db = atan2f(dz, sqrtf(dx * dx + dy * dy));
  float rmax = hmax16(r_dis);
  float pn = sqrtf(px * px + py * py + pz * pz);
  lg = (rmax * rmax * rmax) / (pn * pn * pn);
  if (lane < 16) {
    ldsT[0 * 16 + k] = r_alpha - da;
    ldsT[1 * 16 + k] = r_beta - db;
    ldsT[2 * 16 + k] = r_dis;
    ldsT[3 * 16 + k] = px;
    ldsT[4 * 16 + k] = py;
    ldsT[5 * 16 + k] = pz;
    ldsT[6 * 16 + k] = qx;
    ldsT[7 * 16 + k] = qy;
    ldsT[8 * 16 + k] = qz;
    ldsT[9 * 16 + k] = 0.f;   // pad 9 -> 12 channels
    ldsT[10 * 16 + k] = 0.f;
    ldsT[11 * 16 + k] = 0.f;
  }
  v8f acc = zero8();
#pragma unroll
  for (int c0 = 0; c0 < 12; c0 += 4)
    acc = wmma4(load_A(lm1W, 9, 0, c0, lane), load_B_lds(ldsT, c0, lane), acc);
  int hb = (lane >> 4) * 8;
#pragma unroll
  for (int r = 0; r < 8; ++r) {
    int o = r + hb;
    acc[r] = fmaxf(lm1g[o] * acc[r] + lm1b[o], 0.0f);
  }
  return acc;
}

// Attention pool core: logits = fcW(32x34) @ cat(34x16), softmax over K,
// f_lc[c] = sum_k f_set[c][k]*att[c][k].  cat rows: 0=g_dis,1=2*f_dis,
// 2..33=f_set, 34..35=zero pad.
__device__ __forceinline__ void pool_att(const float* __restrict__ fcW,
                                         const float* cat, float* flc,
                                         int lane) {
  int col = lane & 15;
  int hb = (lane >> 4) * 8;
#pragma unroll
  for (int mt = 0; mt < 2; ++mt) {
    v8f acc = zero8();
#pragma unroll
    for (int c0 = 0; c0 < 36; c0 += 4)
      acc = wmma4(load_A(fcW, 34, mt * 16, c0, lane),
                  load_B_lds(cat, c0, lane), acc);
#pragma unroll
    for (int r = 0; r < 8; ++r) {
      int ch = mt * 16 + r + hb;
      float x = acc[r];
      float mv = hmax16(x);
      float e = __expf(x - mv);
      float s = hsum16(e);
      float att = e / s;
      float w = att * cat[(2 + ch) * 16 + col];
      float fs = hsum16(w);
      if (col == 0) flc[ch] = fs;
    }
  }
}

// ---------------- Kernel 2: lrep, pool-1 -> feat1, lg_ratio ----------------
__global__ void k2_pool1(const float* __restrict__ xyz,
                         const int* __restrict__ nidx,
                         const float* __restrict__ lm1W,
                         const float* __restrict__ lm1g,
                         const float* __restrict__ lm1b,
                         const float* __restrict__ p1fcW,
                         const float* __restrict__ p1mW,
                         const float* __restrict__ p1mg,
                         const float* __restrict__ p1mb,
                         const float* __restrict__ fpc,
                         float* __restrict__ feat1, float* __restrict__ lgout) {
  __shared__ float smem[8 * 800];
  int lane = threadIdx.x & 31;
  int wave = threadIdx.x >> 5;
  float* ldsT = smem + wave * 800; // 12x16
  float* cat = ldsT + 192;         // 36x16
  float* flc = cat + 576;          // 32
  int p = blockIdx.x * 8 + wave;
  int b = p / NN;
  int n = p % NN;
  float gdis, lg;
  int idx;
  v8f lrep = geo_lrep(xyz, nidx, lm1W, lm1g, lm1b, b, n, ldsT, lane, gdis, idx, lg);
  if (lane == 0) lgout[b * NN + n] = lg;
  int k = lane & 15;
  int hb = (lane >> 4) * 8;
#pragma unroll
  for (int r = 0; r < 8; ++r) cat[(18 + r + hb) * 16 + k] = lrep[r];
  __builtin_prefetch(&fpc[(b * 16) * NN + idx], 0, 1);
  float aabs = 0.f;
#pragma unroll
  for (int c = 0; c < 16; ++c) {
    float fn = fpc[(b * 16 + c) * NN + idx];
    float fc = fpc[(b * 16 + c) * NN + n];
    aabs += fabsf(fc - fn);
    if (lane < 16) cat[(2 + c) * 16 + k] = fn;
  }
  float fdis = __expf(-aabs * 0.0625f);
  if (lane < 16) {
    cat[0 * 16 + k] = gdis;
    cat[1 * 16 + k] = 2.0f * fdis;
    cat[34 * 16 + k] = 0.f;
    cat[35 * 16 + k] = 0.f;
  }
  pool_att(p1fcW, cat, flc, lane);
  if (lane < 16) {
    float s = 0.f;
#pragma unroll
    for (int c = 0; c < 32; ++c) s += p1mW[lane * 32 + c] * flc[c];
    feat1[(b * 16 + lane) * NN + n] =
        fmaxf(p1mg[lane] * s + p1mb[lane], 0.f);
  }
}

// ------------- Kernel 3: recompute lrep, lrep2, pool-2 -> flc2 -------------
__global__ void k3_pool2(const float* __restrict__ xyz,
                         const int* __restrict__ nidx,
                         const float* __restrict__ lm1W,
                         const float* __restrict__ lm1g,
                         const float* __restrict__ lm1b,
                         const float* __restrict__ lm2W,
                         const float* __restrict__ lm2g,
                         const float* __restrict__ lm2b,
                         const float* __restrict__ p2fcW,
                         const float* __restrict__ p2mW,
                         const float* __restrict__ p2mg,
                         const float* __restrict__ p2mb,
                         const float* __restrict__ feat1,
                         float* __restrict__ flc2out) {
  __shared__ float smem[8 * 1056];
  int lane = threadIdx.x & 31;
  int wave = threadIdx.x >> 5;
  float* ldsT = smem + wave * 1056; // 12x16
  float* tile16 = ldsT + 192;       // 16x16 (lrep staged for B operand)
  float* cat = tile16 + 256;        // 36x16
  float* flc = cat + 576;           // 32
  int p = blockIdx.x * 8 + wave;
  int b = p / NN;
  int n = p % NN;
  float gdis, lg;
  int idx;
  v8f lrep = geo_lrep(xyz, nidx, lm1W, lm1g, lm1b, b, n, ldsT, lane, gdis, idx, lg);
  int k = lane & 15;
  int hb = (lane >> 4) * 8;
#pragma unroll
  for (int r = 0; r < 8; ++r) tile16[(r + hb) * 16 + k] = lrep[r];
  // lrep2 = cbr(lm2 @ lrep)
  v8f acc = zero8();
#pragma unroll
  for (int c0 = 0; c0 < 16; c0 += 4)
    acc = wmma4(load_A(lm2W, 16, 0, c0, lane), load_B_lds(tile16, c0, lane), acc);
#pragma unroll
  for (int r = 0; r < 8; ++r) {
    int o = r + hb;
    cat[(18 + o) * 16 + k] = fmaxf(lm2g[o] * acc[r] + lm2b[o], 0.f);
  }
  float aabs = 0.f;
#pragma unroll
  for (int c = 0; c < 16; ++c) {
    float fn = feat1[(b * 16 + c) * NN + idx];
    float fc = feat1[(b * 16 + c) * NN + n];
    aabs += fabsf(fc - fn);
    if (lane < 16) cat[(2 + c) * 16 + k] = fn;
  }
  float fdis = __expf(-aabs * 0.0625f);
  if (lane < 16) {
    cat[0 * 16 + k] = gdis;
    cat[1 * 16 + k] = 2.0f * fdis;
    cat[34 * 16 + k] = 0.f;
    cat[35 * 16 + k] = 0.f;
  }
  pool_att(p2fcW, cat, flc, lane);
  { // p2m: 32 outputs, one per lane
    float s = 0.f;
#pragma unroll
    for (int c = 0; c < 32; ++c) s += p2mW[lane * 32 + c] * flc[c];
    flc2out[(b * 32 + lane) * NN + n] =
        fmaxf(p2mg[lane] * s + p2mb[lane], 0.f);
  }
}

// --- Kernel 4: out = cbr(m4 @ [m2@flc2 + sc@feature ; m3@[xyz,lg]]) ---
__global__ void k4_head(const float* __restrict__ feature,
                        const float* __restrict__ xyz,
                        const float* __restrict__ m2W,
                        const float* __restrict__ m2g,
                        const float* __restrict__ m2b,
                        const float* __restrict__ scW,
                        const float* __restrict__ scg,
                        const float* __restrict__ scb,
                        const float* __restrict__ m3W,
                        const float* __restrict__ m3g,
                        const float* __restrict__ m3b,
                        const float* __restrict__ m4W,
                        const float* __restrict__ m4g,
                        const float* __restrict__ m4b,
                        const float* __restrict__ flc2,
                        const float* __restrict__ lg, float* __restrict__ out) {
  __shared__ float smem[4 * 2048];
  int lane = threadIdx.x & 31;
  int wave = threadIdx.x >> 5; // 0..3
  float* H = smem + wave * 2048; // 128 x 16 concat tile
  int tile = blockIdx.x * 4 + wave;
  int b = tile / NT;
  int n0 = (tile % NT) * 16;
  int col = lane & 15;
  int hb = (lane >> 4) * 8;
  int n = n0 + col;
  const float* fb = feature + b * 16 * NN;
  const float* xb = flc2 + b * 32 * NN;
#pragma unroll
  for (int mt = 0; mt < 4; ++mt) {
    v8f a_sc = zero8();
#pragma unroll
    for (int c0 = 0; c0 < 16; c0 += 4)
      a_sc = wmma4(load_A(scW, 16, mt * 16, c0, lane),
                   load_B_chan(fb, c0, n, lane), a_sc);
    v8f a_m2 = zero8();
#pragma unroll
    for (int c0 = 0; c0 < 32; c0 += 4)
      a_m2 = wmma4(load_A(m2W, 32, mt * 16, c0, lane),
                   load_B_chan(xb, c0, n, lane), a_m2);
    v2f bg; // rows 0..3 of [xyz.x, xyz.y, xyz.z, lg_ratio]
    {
      int rb = (lane >> 4) << 1;
      bg.x = (rb == 0) ? xyz[(b * NN + n) * 3 + 0] : xyz[(b * NN + n) * 3 + 2];
      bg.y = (rb == 0) ? xyz[(b * NN + n) * 3 + 1] : lg[b * NN + n];
    }
    v8f a_m3 = wmma4(load_A(m3W, 4, mt * 16, 0, lane), bg, zero8());
#pragma unroll
    for (int r = 0; r < 8; ++r) {
      int o = mt * 16 + r + hb;
      float h1 = (m2g[o] * a_m2[r] + m2b[o]) + (scg[o] * a_sc[r] + scb[o]);
      float h2 = m3g[o] * a_m3[r] + m3b[o];
      H[o * 16 + col] = h1;
      H[(64 + o) * 16 + col] = h2;
    }
  }
#pragma unroll
  for (int mt = 0; mt < 4; ++mt) {
    v8f acc = zero8();
#pragma unroll
    for (int c0 = 0; c0 < 128; c0 += 4)
      acc = wmma4(load_A(m4W, 128, mt * 16, c0, lane),
                  load_B_lds(H, c0, lane), acc);
#pragma unroll
    for (int r = 0; r < 8; ++r) {
      int o = mt * 16 + r + hb;
      out[(b * 64 + o) * NN + n] = fmaxf(m4g[o] * acc[r] + m4b[o], 0.f);
    }
  }
}

extern "C" void kernel_launch(void* const* d_in, const int* in_sizes, int n_in,
                              void* d_out, int out_size, void* d_ws,
                              size_t ws_size, hipStream_t stream) {
  (void)in_sizes; (void)n_in; (void)out_size; (void)ws_size;
  const float* feature = (const float*)d_in[0];
  const float* xyz = (const float*)d_in[1];
  const float* m1W = (const float*)d_in[2];
  const float* m1g = (const float*)d_in[3];
  const float* m1b = (const float*)d_in[4];
  const float* lm1W = (const float*)d_in[5];
  const float* lm1g = (const float*)d_in[6];
  const float* lm1b = (const float*)d_in[7];
  const float* lm2W = (const float*)d_in[8];
  const float* lm2g = (const float*)d_in[9];
  const float* lm2b = (const float*)d_in[10];
  const float* p1fcW = (const float*)d_in[11];
  const float* p1mW = (const float*)d_in[12];
  const float* p1mg = (const float*)d_in[13];
  const float* p1mb = (const float*)d_in[14];
  const float* p2fcW = (const float*)d_in[15];
  const float* p2mW = (const float*)d_in[16];
  const float* p2mg = (const float*)d_in[17];
  const float* p2mb = (const float*)d_in[18];
  const float* m2W = (const float*)d_in[19];
  const float* m2g = (const float*)d_in[20];
  const float* m2b = (const float*)d_in[21];
  const float* scW = (const float*)d_in[22];
  const float* scg = (const float*)d_in[23];
  const float* scb = (const float*)d_in[24];
  const float* m3W = (const float*)d_in[25];
  const float* m3g = (const float*)d_in[26];
  const float* m3b = (const float*)d_in[27];
  const float* m4W = (const float*)d_in[28];
  const float* m4g = (const float*)d_in[29];
  const float* m4b = (const float*)d_in[30];
  const int* nidx = (const int*)d_in[31];

  float* ws = (float*)d_ws;
  float* fpc = ws;                       // B*16*N
  float* feat1 = fpc + BB * 16 * NN;     // B*16*N
  float* flc2 = feat1 + BB * 16 * NN;    // B*32*N
  float* lgr = flc2 + BB * 32 * NN;      // B*N   (total ~21.3 MB)
  float* out = (float*)d_out;

  k1_fpc<<<BB * NT / 8, 256, 0, stream>>>(feature, m1W, m1g, m1b, fpc);
  k2_pool1<<<BB * NN / 8, 256, 0, stream>>>(xyz, nidx, lm1W, lm1g, lm1b, p1fcW,
                                            p1mW, p1mg, p1mb, fpc, feat1, lgr);
  k3_pool2<<<BB * NN / 8, 256, 0, stream>>>(xyz, nidx, lm1W, lm1g, lm1b, lm2W,
                                            lm2g, lm2b, p2fcW, p2mW, p2mg,
                                            p2mb, feat1, flc2);
  k4_head<<<BB * NT / 4, 128, 0, stream>>>(feature, xyz, m2W, m2g, m2b, scW,
                                           scg, scb, m3W, m3g, m3b, m4W, m4g,
                                           m4b, flc2, lgr, out);
}